// GraphConvolution_29059748725486
// MI455X (gfx1250) — compile-verified
//
#include <hip/hip_runtime.h>

typedef __attribute__((ext_vector_type(2))) float v2f;
typedef __attribute__((ext_vector_type(8))) float v8f;

#define D 128          // D_IN == D_OUT == 128
#define EPB 32         // edges per block in scatter stage

// ---------------------------------------------------------------------------
// Stage 1: support = x @ W + b  using V_WMMA_F32_16X16X4_F32 (exact f32 path).
// Block = 256 threads = 8 wave32. Block covers 16 rows of x; wave w covers
// output columns [16w, 16w+16). K loop: 32 steps of K=4.
// ---------------------------------------------------------------------------
__global__ __launch_bounds__(256) void gcn_gemm_wmma(
    const float* __restrict__ x, const float* __restrict__ W,
    const float* __restrict__ bias, float* __restrict__ support, int N)
{
    // 16 x 128 f32 tile of x; pad row stride to 132 dwords so the 16 lanes
    // reading one K-column hit banks (4m + k) mod 64 -> conflict-free.
    __shared__ float xs[16][132];

    const int tid  = threadIdx.x;
    const int row0 = blockIdx.x * 16;

    // Stage the x tile (clamp rows for a partial last tile; N=100000 is exact).
    for (int i = tid; i < 16 * D; i += 256) {
        int r = i >> 7, c = i & 127;
        int rr = row0 + r; if (rr >= N) rr = N - 1;
        xs[r][c] = x[(size_t)rr * D + c];
    }
    __syncthreads();

    const int wave = tid >> 5;     // 0..7 -> N-slab
    const int lane = tid & 31;
    const int col0 = wave * 16;
    const int m    = lane & 15;    // A: row M ; B/C/D: column N
    const int hi   = lane >> 4;    // half-wave selects K pair / M+8

    v8f acc = {};
    #pragma unroll 4
    for (int k4 = 0; k4 < D / 4; ++k4) {
        const int kb = k4 * 4 + 2 * hi;
        // A 16x4 f32: lane m holds K = kb, kb+1 in VGPR0/1 (ds_load_b64)
        v2f a;
        a.x = xs[m][kb + 0];
        a.y = xs[m][kb + 1];
        // B 4x16 f32: VGPR v holds row K = kb+v, column col0+m (row-major W)
        v2f bv;
        bv.x = W[(size_t)(kb + 0) * D + col0 + m];
        bv.y = W[(size_t)(kb + 1) * D + col0 + m];
        // 8-arg form: (neg_a, A, neg_b, B, c_mod, C, reuse_a, reuse_b)
        acc = __builtin_amdgcn_wmma_f32_16x16x4_f32(
            false, a, false, bv, (short)0, acc, false, false);
    }

    const float bi = bias[col0 + m];
    // C/D layout: VGPR r -> M = r + 8*hi, N = m
    const int mrow0 = row0 + 8 * hi;
    float* outp = support + (size_t)mrow0 * D + col0 + m;
    #pragma unroll
    for (int r = 0; r < 8; ++r) {
        if (mrow0 + r < N) outp[(size_t)r * D] = acc[r] + bi;
    }
}

// ---------------------------------------------------------------------------
// Stage 2: out[dst] += w_e * support[src].  edge_dst is SORTED, so each
// thread (one per feature) accumulates a run of identical-dst edges in a
// register and flushes with one global_atomic_add_f32 per run per feature.
// support (51.2 MB) is L2-resident, so gathers are L2-bandwidth traffic.
// ---------------------------------------------------------------------------
__global__ __launch_bounds__(128) void gcn_scatter(
    const float* __restrict__ support, const float* __restrict__ ew,
    const int* __restrict__ esrc, const int* __restrict__ edst,
    float* __restrict__ out, int E)
{
    const int f = threadIdx.x;                 // feature 0..127
    long e   = (long)blockIdx.x * EPB;
    long end = e + EPB; if (end > E) end = E;

    float acc = 0.0f;
    int cur = -1;
    for (; e < end; ++e) {
        const int   s = esrc[e];
        const int   d = edst[e];
        const float w = ew[e];
        if (d != cur) {
            if (cur >= 0) atomicAdd(&out[(size_t)cur * D + f], acc);
            cur = d;
            acc = 0.0f;
        }
        acc = fmaf(w, support[(size_t)s * D + f], acc);
    }
    if (cur >= 0) atomicAdd(&out[(size_t)cur * D + f], acc);
}

// ---------------------------------------------------------------------------
extern "C" void kernel_launch(void* const* d_in, const int* in_sizes, int n_in,
                              void* d_out, int out_size, void* d_ws, size_t ws_size,
                              hipStream_t stream)
{
    const float* x    = (const float*)d_in[0];   // [N, 128]
    const float* W    = (const float*)d_in[1];   // [128, 128]
    const float* b    = (const float*)d_in[2];   // [128]
    const float* ew   = (const float*)d_in[3];   // [E]
    const int*   esrc = (const int*)  d_in[4];   // [E]
    const int*   edst = (const int*)  d_in[5];   // [E] sorted
    float*       out  = (float*)d_out;           // [N, 128]

    const int N = in_sizes[0] / D;
    const int E = in_sizes[3];

    float* support = (float*)d_ws;               // N*D*4 = 51.2 MB scratch

    hipMemsetAsync(d_out, 0, (size_t)out_size * sizeof(float), stream);

    const int gridA = (N + 15) / 16;             // 6250 blocks
    gcn_gemm_wmma<<<gridA, 256, 0, stream>>>(x, W, b, support, N);

    const int gridB = (E + EPB - 1) / EPB;       // 100000 blocks
    gcn_scatter<<<gridB, 128, 0, stream>>>(support, ew, esrc, edst, out, E);
}